// GATConv_54692113547387
// MI455X (gfx1250) — compile-verified
//
#include <hip/hip_runtime.h>
#include <hip/hip_bf16.h>

#define NHEADS 4
#define HDIM   32
#define NNODES 1024
#define BATCH  8
#define INDIM  128
#define OUTDIM 128
#define NEG_INF_F (-1000000000.0f)

typedef float v2f __attribute__((ext_vector_type(2)));
typedef float v8f __attribute__((ext_vector_type(8)));

// ---------------------------------------------------------------------------
// Kernel 1: h = x @ W^T + b    (x: [8192,128] fp32, W: [128,128], h: [8192,128])
// One block = 16 output rows; 8 waves each own a 16-wide column tile.
// fp32 WMMA 16x16x4, K-loop of 32.
// ---------------------------------------------------------------------------
__global__ __launch_bounds__(256) void gat_linear(const float* __restrict__ x,
                                                  const float* __restrict__ W,
                                                  const float* __restrict__ bias,
                                                  float* __restrict__ h)
{
    const int lane    = threadIdx.x & 31;
    const int wave    = threadIdx.x >> 5;     // column tile 0..7
    const int rowTile = blockIdx.x;           // 0..511
    const int rloc    = lane & 15;            // M = lane%16 for A frag
    const int kq      = (lane >> 4) * 2;      // lane<16 -> K {0,1}; lane>=16 -> K {2,3}
    const int col     = wave * 16 + rloc;     // N = lane%16 for B/C frags

    // Stage the 16x128 x-tile in LDS (pad 132 -> bank stride 4, conflict-free)
    __shared__ float xs[16][132];
    for (int i = threadIdx.x; i < 16 * 128; i += 256) {
        int rr = i >> 7, cc = i & 127;
        xs[rr][cc] = x[(rowTile * 16 + rr) * INDIM + cc];
    }
    __syncthreads();

    v8f acc = {};
    for (int kk = 0; kk < INDIM; kk += 4) {
        v2f a, b;
        a.x = xs[rloc][kk + kq + 0];
        a.y = xs[rloc][kk + kq + 1];
        // B[k][n] = W[n][k]  (h = x W^T)
        b.x = W[col * INDIM + kk + kq + 0];
        b.y = W[col * INDIM + kk + kq + 1];
        acc = __builtin_amdgcn_wmma_f32_16x16x4_f32(false, a, false, b,
                                                    (short)0, acc, false, false);
    }

    const float bv = bias[col];
    const int rbase = rowTile * 16 + ((lane >> 4) << 3); // M = v + 8*(lane>=16)
#pragma unroll
    for (int v = 0; v < 8; ++v)
        h[(rbase + v) * OUTDIM + col] = acc[v] + bv;
}

// ---------------------------------------------------------------------------
// Kernel 2: s[b,h,n] = h[b,n,h,:]·a[:32],  t[b,h,n] = h[b,n,h,:]·a[32:]
// ---------------------------------------------------------------------------
__global__ __launch_bounds__(256) void gat_scores(const float* __restrict__ h,
                                                  const float* __restrict__ a,
                                                  float* __restrict__ s,
                                                  float* __restrict__ t)
{
    int idx = blockIdx.x * blockDim.x + threadIdx.x;   // b*4096 + head*1024 + n
    if (idx >= BATCH * NHEADS * NNODES) return;
    int n    = idx & (NNODES - 1);
    int head = (idx >> 10) & (NHEADS - 1);
    int b    = idx >> 12;
    const float* hp = h + ((size_t)(b * NNODES + n) * OUTDIM + head * HDIM);
    float ss = 0.f, tt = 0.f;
#pragma unroll
    for (int d = 0; d < HDIM; ++d) {
        float hv = hp[d];
        ss += hv * a[d];
        tt += hv * a[HDIM + d];
    }
    s[idx] = ss;
    t[idx] = tt;
}

// ---------------------------------------------------------------------------
// Kernel 3: masked leaky-relu softmax over j, then out_tile = (P @ H_head) / l.
// Block = (b, head, 16-row tile); 4 waves, each owns 256 columns.
// Pass 1: online (m,l) per lane, merged in LDS. Pass 2: fp32 WMMA P·H.
// ---------------------------------------------------------------------------
__global__ __launch_bounds__(128) void gat_attn(const float* __restrict__ h,
                                                const int*   __restrict__ adj,
                                                const float* __restrict__ s,
                                                const float* __restrict__ t,
                                                float* __restrict__ out)
{
    const int rowTile = blockIdx.x & 63;
    const int head    = (blockIdx.x >> 6) & (NHEADS - 1);
    const int b       = blockIdx.x >> 8;

    const int tid  = threadIdx.x;
    const int wave = tid >> 5;
    const int lane = tid & 31;
    const int r    = lane & 15;   // local row (A-frag M = lane%16)
    const int jl   = lane >> 4;   // which K pair this lane feeds

    __shared__ float pm[128], pl[128];
    __shared__ float gm[16], gl[16];
    __shared__ float pc[4][2][8][32];

    const int   iGlob  = rowTile * 16 + r;
    const float s_row  = s[(b * NHEADS + head) * NNODES + iGlob];
    const float* t_bh  = t + (b * NHEADS + head) * NNODES;
    const int*  adj_row = adj + (size_t)(b * NNODES + iGlob) * NNODES;

    // ---- pass 1: online row max / sum(exp) over this wave's 256 columns ----
    const int jbase = wave * 256;
    float m = -__builtin_inff(), l = 0.f;
    for (int it = 0; it < 128; ++it) {
        int j = jbase + jl + it * 2;
        float e = s_row + t_bh[j];
        e = (e > 0.f) ? e : 0.2f * e;                 // leaky_relu(0.2)
        e = (adj_row[j] == 0) ? NEG_INF_F : e;        // mask
        float mn = fmaxf(m, e);
        l = l * __expf(m - mn) + __expf(e - mn);
        m = mn;
    }
    pm[tid] = m; pl[tid] = l;
    __syncthreads();
    if (tid < 16) {                                   // merge 8 partials per row
        float gmv = -__builtin_inff();
        for (int q = 0; q < 8; ++q) {
            int src = (q >> 1) * 32 + (q & 1) * 16 + tid;
            gmv = fmaxf(gmv, pm[src]);
        }
        float glv = 0.f;
        for (int q = 0; q < 8; ++q) {
            int src = (q >> 1) * 32 + (q & 1) * 16 + tid;
            glv += pl[src] * __expf(pm[src] - gmv);
        }
        gm[tid] = gmv;
        gl[tid] = glv;
    }
    __syncthreads();

    // ---- pass 2: accumulate exp(e - m_row) @ H_head with fp32 WMMA ----
    const float mrow = gm[r];
    const float* hb = h + (size_t)b * NNODES * OUTDIM + head * HDIM;
    const int n0 = lane & 15;
    v8f c0 = {}, c1 = {};
    for (int it = 0; it < 64; ++it) {
        int jA = jbase + it * 4 + jl * 2;             // K = jl*2 {+0,+1}
        v2f a;
        {
            float e = s_row + t_bh[jA];
            e = (e > 0.f) ? e : 0.2f * e;
            e = (adj_row[jA] == 0) ? NEG_INF_F : e;
            a.x = __expf(e - mrow);
            e = s_row + t_bh[jA + 1];
            e = (e > 0.f) ? e : 0.2f * e;
            e = (adj_row[jA + 1] == 0) ? NEG_INF_F : e;
            a.y = __expf(e - mrow);
        }
        v2f b0, b1;                                   // B[k][n] = h[j, head*32 + ct*16 + n]
        b0.x = hb[(size_t)(jA + 0) * OUTDIM + n0];
        b0.y = hb[(size_t)(jA + 1) * OUTDIM + n0];
        b1.x = hb[(size_t)(jA + 0) * OUTDIM + 16 + n0];
        b1.y = hb[(size_t)(jA + 1) * OUTDIM + 16 + n0];
        c0 = __builtin_amdgcn_wmma_f32_16x16x4_f32(false, a, false, b0,
                                                   (short)0, c0, false, false);
        c1 = __builtin_amdgcn_wmma_f32_16x16x4_f32(false, a, false, b1,
                                                   (short)0, c1, false, false);
    }
#pragma unroll
    for (int v = 0; v < 8; ++v) {
        pc[wave][0][v][lane] = c0[v];
        pc[wave][1][v][lane] = c1[v];
    }
    __syncthreads();

    // ---- reduce partials across waves, normalize by l, store ----
    for (int idx = tid; idx < 512; idx += 128) {
        int ct = idx >> 8;
        int v  = (idx >> 5) & 7;
        int ln = idx & 31;
        float sum = pc[0][ct][v][ln] + pc[1][ct][v][ln] +
                    pc[2][ct][v][ln] + pc[3][ct][v][ln];
        int rowL = v + ((ln >> 4) << 3);              // D layout: M = v + 8*(lane>=16)
        int col  = ct * 16 + (ln & 15);
        out[(size_t)(b * NNODES + rowTile * 16 + rowL) * OUTDIM + head * HDIM + col]
            = sum / gl[rowL];
    }
}

// ---------------------------------------------------------------------------
extern "C" void kernel_launch(void* const* d_in, const int* in_sizes, int n_in,
                              void* d_out, int out_size, void* d_ws, size_t ws_size,
                              hipStream_t stream)
{
    const float* x    = (const float*)d_in[0];
    const int*   adj  = (const int*)  d_in[1];
    const float* W    = (const float*)d_in[2];
    const float* bias = (const float*)d_in[3];
    const float* a    = (const float*)d_in[4];
    float* out = (float*)d_out;

    float* h = (float*)d_ws;                          // 8192*128 fp32 = 4 MB
    float* s = h + BATCH * NNODES * OUTDIM;           // 32768 fp32
    float* t = s + BATCH * NHEADS * NNODES;           // 32768 fp32

    gat_linear<<<(BATCH * NNODES) / 16, 256, 0, stream>>>(x, W, bias, h);
    gat_scores<<<(BATCH * NHEADS * NNODES) / 256, 256, 0, stream>>>(h, a, s, t);
    gat_attn<<<BATCH * NHEADS * (NNODES / 16), 128, 0, stream>>>(h, adj, s, t, out);
}